// SpikingRON_86028194939427
// MI455X (gfx1250) — compile-verified
//
#include <hip/hip_runtime.h>

// Problem dims
#define B_      64
#define T_      512
#define NI      64
#define NH      1024
#define THARM   256
#define TSPIKE  256
#define DT_     0.042f
#define THRESH  0.008f
#define RESETV  0.001f

// Output layout (flat fp32 concat): hys | hzs | us | spikes
#define HYS_SZ  ((size_t)B_ * T_ * NH)          // 33554432
#define US_OFF  (2 * HYS_SZ)                    // 67108864
#define SP_OFF  (US_OFF + (size_t)B_ * TSPIKE * NH)

// LDS strides padded so WMMA A-operand ds loads are bank-conflict free
#define HSTRIDE 1028
#define XSTRIDE 68

typedef float v2f __attribute__((ext_vector_type(2)));
typedef float v8f __attribute__((ext_vector_type(8)));

static __device__ __forceinline__ v8f wmma4(v2f a, v2f b, v8f c) {
  // D = A(16x4 f32) * B(4x16 f32) + C(16x16 f32)
  return __builtin_amdgcn_wmma_f32_16x16x4_f32(false, a, false, b, (short)0, c,
                                               false, false);
}

// ---- pack h2h into per-lane WMMA-B order: pB[(tile*256+ks)*32+lane] = float2
//      {h2h[k][n], h2h[k+1][n]}, n = tile*16 + lane%16, k = 4*ks + 2*(lane/16)
__global__ void pack_h2h(const float* __restrict__ h2h, float* __restrict__ pB) {
  int idx = blockIdx.x * blockDim.x + threadIdx.x;
  if (idx >= 64 * 256 * 32) return;
  int lane = idx & 31;
  int ks   = (idx >> 5) & 255;
  int tile = idx >> 13;
  int n = tile * 16 + (lane & 15);
  int k = ks * 4 + 2 * (lane >> 4);
  v2f v;
  v.x = h2h[(size_t)k * NH + n];
  v.y = h2h[(size_t)(k + 1) * NH + n];
  ((v2f*)pB)[idx] = v;
}

__global__ void pack_x2h(const float* __restrict__ x2h, float* __restrict__ pX) {
  int idx = blockIdx.x * blockDim.x + threadIdx.x;
  if (idx >= 64 * 16 * 32) return;
  int lane = idx & 31;
  int ks   = (idx >> 5) & 15;
  int tile = idx >> 9;
  int n = tile * 16 + (lane & 15);
  int k = ks * 4 + 2 * (lane >> 4);
  v2f v;
  v.x = x2h[(size_t)k * NH + n];
  v.y = x2h[(size_t)(k + 1) * NH + n];
  ((v2f*)pX)[idx] = v;
}

// ---- one-shot parallel GEMM: xw[(b*T+t)][n] = x[b][t][:] @ x2h[:,n]
//      rows = b*T+t (32768), cols = 1024, K = 64. 8 waves/block, 1 tile/wave.
__global__ __launch_bounds__(256)
void xw_gemm(const float* __restrict__ x, const float* __restrict__ pX,
             float* __restrict__ xw) {
  const int lane = threadIdx.x & 31;
  const int wave = threadIdx.x >> 5;
  const int tile = blockIdx.x * 8 + wave;   // 131072 tiles = 2048 mt * 64 nt
  const int mt = tile >> 6;
  const int nt = tile & 63;
  const int lmod = lane & 15, lhi = lane >> 4;
  const int row0 = mt * 16;

  const v2f* bp = (const v2f*)pX + (size_t)(nt * 16) * 32 + lane;
  const float* ap = x + (size_t)(row0 + lmod) * NI + 2 * lhi;

  v8f acc = {0, 0, 0, 0, 0, 0, 0, 0};
#pragma unroll
  for (int kk = 0; kk < 16; ++kk)
    acc = wmma4(*(const v2f*)(ap + 4 * kk), bp[kk * 32], acc);

  const int n = nt * 16 + lmod;
#pragma unroll
  for (int v = 0; v < 8; ++v) {
    const int m = v + lhi * 8;
    xw[(size_t)(row0 + m) * NH + n] = acc[v];
  }
}

// ---- main recurrent kernel: 1 block per 16-batch slice, 1024 threads = 32 waves,
//      wave w owns n-tiles {2w, 2w+1} (each 16 columns).
//      USE_XW: x-contribution precomputed in xw (no x-GEMM on the serial path).
template <bool USE_XW>
__global__ __launch_bounds__(1024, 1)
void ron_main(const float* __restrict__ x, const float* __restrict__ bias,
              const float* __restrict__ gamma, const float* __restrict__ eps,
              const float* __restrict__ pB, const float* __restrict__ pX,
              const float* __restrict__ xw, float* __restrict__ out) {
  __shared__ float hy_s[16 * HSTRIDE];
  __shared__ float x_s[16 * XSTRIDE];

  const int tid  = threadIdx.x;
  const int lane = tid & 31;
  const int wave = tid >> 5;
  const int b0   = blockIdx.x * 16;
  const int tile0 = wave * 2;
  const int lmod = lane & 15;
  const int lhi  = lane >> 4;

  const int n0 = tile0 * 16 + lmod;
  const int n1 = n0 + 16;
  const float bia0 = bias[n0],  bia1 = bias[n1];
  const float g0   = gamma[n0], g1   = gamma[n1];
  const float e0   = eps[n0],   e1   = eps[n1];

  const v2f* pB0 = (const v2f*)pB + (size_t)(tile0 * 256) * 32 + lane;
  const v2f* pB1 = pB0 + 256 * 32;
  const float* aH = &hy_s[lmod * HSTRIDE + 2 * lhi];
  const float* aX = &x_s[lmod * XSTRIDE + 2 * lhi];
  // volatile views defeat LICM hoist+spill of loop-invariant x2h operands
  const volatile float* vX0 = (const volatile float*)((const v2f*)pX + (size_t)(tile0 * 16) * 32 + lane);
  const volatile float* vX1 = vX0 + 16 * 32 * 2;

  float hz0r[8], hz1r[8];
#pragma unroll
  for (int v = 0; v < 8; ++v) { hz0r[v] = hz1r[v] = 0.f; }

  for (int i = tid; i < 16 * HSTRIDE; i += 1024) hy_s[i] = 0.f;
  __syncthreads();

  const int xm = tid >> 6, xk = tid & 63;   // cooperative x-tile staging role

  // ================= harmonic phase =================
  for (int t = 0; t < THARM; ++t) {
    float xv0[8], xv1[8];
    if (USE_XW) {
#pragma unroll
      for (int v = 0; v < 8; ++v) {
        const size_t o = ((size_t)(b0 + v + lhi * 8) * T_ + t) * NH;
        xv0[v] = xw[o + n0];
        xv1[v] = xw[o + n1];
      }
      if (t + 1 < THARM)
        __builtin_prefetch(&xw[((size_t)(b0 + lmod) * T_ + t + 1) * NH + n0], 0, 1);
    } else {
      x_s[xm * XSTRIDE + xk] = x[((size_t)(b0 + xm) * T_ + t) * NI + xk];
    }
    __syncthreads();                         // hy_s(t-1) [+ x_s] visible

    v8f acc0 = {0, 0, 0, 0, 0, 0, 0, 0};
    v8f acc1 = {0, 0, 0, 0, 0, 0, 0, 0};
    if (!USE_XW) {
#pragma unroll
      for (int kk = 0; kk < 16; ++kk) {      // x @ x2h, K=64
        v2f a = *(const v2f*)(aX + 4 * kk);
        v2f b0v, b1v;
        b0v.x = vX0[kk * 64];  b0v.y = vX0[kk * 64 + 1];
        b1v.x = vX1[kk * 64];  b1v.y = vX1[kk * 64 + 1];
        acc0 = wmma4(a, b0v, acc0);
        acc1 = wmma4(a, b1v, acc1);
      }
    }
#pragma unroll 4
    for (int ks = 0; ks < 256; ++ks) {       // hy @ h2h, K=1024
      v2f a = *(const v2f*)(aH + 4 * ks);
      acc0 = wmma4(a, pB0[ks * 32], acc0);
      acc1 = wmma4(a, pB1[ks * 32], acc1);
    }
    __syncthreads();                         // all hy_s reads done

#pragma unroll
    for (int v = 0; v < 8; ++v) {
      const int m = v + lhi * 8;
      const float hy0 = hy_s[m * HSTRIDE + n0];
      const float hy1 = hy_s[m * HSTRIDE + n1];
      const float pre0 = acc0[v] + (USE_XW ? xv0[v] : 0.f) + bia0;
      const float pre1 = acc1[v] + (USE_XW ? xv1[v] : 0.f) + bia1;
      const float act0 = tanhf(pre0);
      const float act1 = tanhf(pre1);
      hz0r[v] += DT_ * (act0 - g0 * hy0 - e0 * hz0r[v]);
      hz1r[v] += DT_ * (act1 - g1 * hy1 - e1 * hz1r[v]);
      const float ny0 = hy0 + DT_ * hz0r[v];
      const float ny1 = hy1 + DT_ * hz1r[v];
      hy_s[m * HSTRIDE + n0] = ny0;
      hy_s[m * HSTRIDE + n1] = ny1;
      const size_t o = ((size_t)(b0 + m) * T_ + t) * NH;
      __builtin_nontemporal_store(ny0, &out[o + n0]);
      __builtin_nontemporal_store(ny1, &out[o + n1]);
      __builtin_nontemporal_store(hz0r[v], &out[HYS_SZ + o + n0]);
      __builtin_nontemporal_store(hz1r[v], &out[HYS_SZ + o + n1]);
    }
  }
  __syncthreads();                           // final hy_s visible

  // ================= hyW = hy @ h2h (frozen) =================
  v8f w0 = {0, 0, 0, 0, 0, 0, 0, 0};
  v8f w1 = {0, 0, 0, 0, 0, 0, 0, 0};
#pragma unroll 4
  for (int ks = 0; ks < 256; ++ks) {
    v2f a = *(const v2f*)(aH + 4 * ks);
    w0 = wmma4(a, pB0[ks * 32], w0);
    w1 = wmma4(a, pB1[ks * 32], w1);
  }

  // ================= spiking phase =================
  float u0r[8], u1r[8];
#pragma unroll
  for (int v = 0; v < 8; ++v) { u0r[v] = u1r[v] = 0.f; }

  for (int t = 0; t < TSPIKE; ++t) {
    v8f acc0 = {0, 0, 0, 0, 0, 0, 0, 0};
    v8f acc1 = {0, 0, 0, 0, 0, 0, 0, 0};
    if (!USE_XW) {
      __syncthreads();                       // previous x_s readers done
      x_s[xm * XSTRIDE + xk] = x[((size_t)(b0 + xm) * T_ + t) * NI + xk];
      __syncthreads();
#pragma unroll
      for (int kk = 0; kk < 16; ++kk) {      // x @ x2h
        v2f a = *(const v2f*)(aX + 4 * kk);
        v2f b0v, b1v;
        b0v.x = vX0[kk * 64];  b0v.y = vX0[kk * 64 + 1];
        b1v.x = vX1[kk * 64];  b1v.y = vX1[kk * 64 + 1];
        acc0 = wmma4(a, b0v, acc0);
        acc1 = wmma4(a, b1v, acc1);
      }
    }

#pragma unroll
    for (int v = 0; v < 8; ++v) {
      const int m = v + lhi * 8;
      float xa0, xa1;
      if (USE_XW) {
        const size_t ox = ((size_t)(b0 + m) * T_ + t) * NH;
        xa0 = xw[ox + n0];
        xa1 = xw[ox + n1];
      } else {
        xa0 = acc0[v];
        xa1 = acc1[v];
      }
      const float s0 = (u0r[v] > THRESH) ? 1.f : 0.f;
      const float s1 = (u1r[v] > THRESH) ? 1.f : 0.f;
      float uu0 = (s0 != 0.f) ? RESETV : u0r[v];
      float uu1 = (s1 != 0.f) ? RESETV : u1r[v];
      uu0 = uu0 + (-uu0 + w0[v] + xa0) * 0.025f * DT_;   // (R*C) then *DT
      uu1 = uu1 + (-uu1 + w1[v] + xa1) * 0.025f * DT_;
      u0r[v] = uu0;
      u1r[v] = uu1;
      const size_t o = ((size_t)(b0 + m) * TSPIKE + t) * NH;
      __builtin_nontemporal_store(uu0, &out[US_OFF + o + n0]);
      __builtin_nontemporal_store(uu1, &out[US_OFF + o + n1]);
      __builtin_nontemporal_store(s0, &out[SP_OFF + o + n0]);
      __builtin_nontemporal_store(s1, &out[SP_OFF + o + n1]);
    }
  }
}

// ---- broadcast frozen hy/hz over t in [256,512): massively parallel tail fill
__global__ void bcast_tail(float* __restrict__ out) {
  size_t idx = (size_t)blockIdx.x * blockDim.x + threadIdx.x;
  const size_t n = (size_t)B_ * TSPIKE * NH;
  if (idx >= n) return;
  int h = (int)(idx % NH);
  size_t bt = idx / NH;
  int t2 = (int)(bt % TSPIKE) + THARM;
  int b  = (int)(bt / TSPIKE);
  size_t src = ((size_t)b * T_ + (THARM - 1)) * NH + h;
  size_t dst = ((size_t)b * T_ + t2) * NH + h;
  float hy = out[src];
  float hz = out[HYS_SZ + src];
  __builtin_nontemporal_store(hy, &out[dst]);
  __builtin_nontemporal_store(hz, &out[HYS_SZ + dst]);
}

extern "C" void kernel_launch(void* const* d_in, const int* in_sizes, int n_in,
                              void* d_out, int out_size, void* d_ws, size_t ws_size,
                              hipStream_t stream) {
  const float* x     = (const float*)d_in[0];
  const float* x2h   = (const float*)d_in[1];
  const float* h2h   = (const float*)d_in[2];
  const float* bias  = (const float*)d_in[3];
  const float* gamma = (const float*)d_in[4];
  const float* eps   = (const float*)d_in[5];
  float* out = (float*)d_out;

  float* pB = (float*)d_ws;                    // 64*256*32 float2 = 4 MB
  float* pX = pB + (size_t)64 * 256 * 32 * 2;  // 64*16*32 float2 = 256 KB
  float* xw = pX + (size_t)64 * 16 * 32 * 2;   // B*T*NH f32 = 128 MB

  const size_t need_xw =
      (4u << 20) + (256u << 10) + (size_t)B_ * T_ * NH * sizeof(float);
  const bool use_xw = ws_size >= need_xw;

  pack_h2h<<<(64 * 256 * 32 + 255) / 256, 256, 0, stream>>>(h2h, pB);
  pack_x2h<<<(64 * 16 * 32 + 255) / 256, 256, 0, stream>>>(x2h, pX);
  if (use_xw) {
    xw_gemm<<<16384, 256, 0, stream>>>(x, pX, xw);
    ron_main<true><<<4, 1024, 0, stream>>>(x, bias, gamma, eps, pB, pX, xw, out);
  } else {
    ron_main<false><<<4, 1024, 0, stream>>>(x, bias, gamma, eps, pB, pX, xw, out);
  }
  bcast_tail<<<((unsigned)((size_t)B_ * TSPIKE * NH + 255) / 256), 256, 0, stream>>>(out);
}